// AgentAttention_31782757990433
// MI455X (gfx1250) — compile-verified
//
#include <hip/hip_runtime.h>
#include <cmath>

typedef __attribute__((ext_vector_type(2))) float v2f;
typedef __attribute__((ext_vector_type(8))) float v8f;
typedef __attribute__((ext_vector_type(4))) int   v4i;

#define DIMC   768
#define AGENTS 4
#define BATCH  128
#define NTOK   196
#define MROWS  (BATCH * NTOK)          // 25088
#define QKVC   (3 * DIMC)              // 2304
#define POOL1C 384
#define POOL2C 3072
#define ATTN_SCALE 0.03608439182435161f   // 768^-0.5

// -------- CDNA5 async global->LDS copy (ASYNCcnt path), with fallback -------
#if __has_builtin(__builtin_amdgcn_global_load_async_to_lds_b128)
#define HAVE_ASYNC_LDS 1
#else
#define HAVE_ASYNC_LDS 0
#endif

#if HAVE_ASYNC_LDS
typedef __attribute__((address_space(1))) v4i gas_v4i;   // global int4
typedef __attribute__((address_space(3))) v4i las_v4i;   // LDS int4

__device__ __forceinline__ void async_copy16(const float* g, float* l)
{
    __builtin_amdgcn_global_load_async_to_lds_b128(
        (gas_v4i*)(g), (las_v4i*)(l), /*imm offset*/ 0, /*cpol*/ 0);
}
__device__ __forceinline__ void wait_async_lds()
{
#if __has_builtin(__builtin_amdgcn_s_wait_asynccnt)
    __builtin_amdgcn_s_wait_asynccnt(0);
#else
    asm volatile("s_wait_asynccnt 0x0" ::: "memory");
#endif
}
#endif

// ---------------------------------------------------------------------------
// Tiled fp32 GEMM, C = act(A @ B + bias), using V_WMMA_F32_16X16X4_F32.
// Block: 256 threads (8 waves). Block tile 64(M) x 128(N), K staged 16 at a
// time into double-buffered LDS via GLOBAL_LOAD_ASYNC_TO_LDS_B128 (copies of
// chunk k+1 overlap the 16 WMMAs of chunk k). Each wave owns a 32x32 tile as
// 2x2 fp32 WMMA accumulators.
// Requires: M % 64 == 0, N % 128 == 0, K % 16 == 0, lda/ldb % 4 == 0.
// ---------------------------------------------------------------------------
__global__ __launch_bounds__(256)
void gemm_f32_wmma(const float* __restrict__ A, int lda,
                   const float* __restrict__ Bw, int ldb,
                   const float* __restrict__ bias,
                   float* __restrict__ C, int ldc,
                   int M, int N, int K, int act)
{
    __shared__ float As[2][64 * 16];    // 64 rows x 16 k, double buffered
    __shared__ float Bs[2][16 * 128];   // 16 k x 128 cols, double buffered

    const int tid  = threadIdx.x;
    const int lane = tid & 31;
    const int w    = tid >> 5;            // wave 0..7
    const int wm   = (w >> 2) * 32;       // wave M offset in block tile
    const int wn   = (w & 3) * 32;        // wave N offset in block tile
    const int half = lane >> 4;           // 0: lanes 0-15, 1: lanes 16-31
    const int ln   = lane & 15;

    const int m0 = blockIdx.y * 64;
    const int n0 = blockIdx.x * 128;

    v8f acc[2][2] = {};

    // global->LDS staging assignments
    const int arow = tid >> 2;            // 0..63
    const int acol = (tid & 3) * 4;       // 0,4,8,12
    const int brow = tid >> 4;            // 0..15
    const int bcol = (tid & 15) * 4;      // 0..60

    const int nch = K / 16;

    auto stage = [&](int kc, int buf) {
        const int k0 = kc * 16;
        const float* ag  = &A[(size_t)(m0 + arow) * lda + k0 + acol];
        const float* bg0 = &Bw[(size_t)(k0 + brow) * ldb + n0 + bcol];
        const float* bg1 = bg0 + 64;
        float* al  = &As[buf][arow * 16 + acol];
        float* bl0 = &Bs[buf][brow * 128 + bcol];
        float* bl1 = bl0 + 64;
#if HAVE_ASYNC_LDS
        async_copy16(ag,  al);
        async_copy16(bg0, bl0);
        async_copy16(bg1, bl1);
#else
        *reinterpret_cast<float4*>(al)  = *reinterpret_cast<const float4*>(ag);
        *reinterpret_cast<float4*>(bl0) = *reinterpret_cast<const float4*>(bg0);
        *reinterpret_cast<float4*>(bl1) = *reinterpret_cast<const float4*>(bg1);
        __builtin_prefetch(ag + 16, 0, 1);                       // next K-chunk
        __builtin_prefetch(bg0 + 16 * (size_t)ldb, 0, 1);
#endif
    };

    stage(0, 0);
    for (int kc = 0; kc < nch; ++kc) {
        const int buf = kc & 1;
#if HAVE_ASYNC_LDS
        wait_async_lds();                 // own chunk-kc async copies complete
#endif
        __syncthreads();                  // all waves' copies visible in LDS
        if (kc + 1 < nch)
            stage(kc + 1, buf ^ 1);       // overlap next copies with WMMA work

        const float* Asb = As[buf];
        const float* Bsb = Bs[buf];
        #pragma unroll
        for (int kk = 0; kk < 4; ++kk) {
            // ISA layout: lanes 0-15 hold K = {kb, kb+1}, lanes 16-31 {kb+2, kb+3}
            const int kb = kk * 4 + 2 * half;
            v2f afrag[2], bfrag[2];
            #pragma unroll
            for (int tm = 0; tm < 2; ++tm) {
                afrag[tm] = *reinterpret_cast<const v2f*>(
                    &Asb[(wm + tm * 16 + ln) * 16 + kb]);
            }
            #pragma unroll
            for (int tn = 0; tn < 2; ++tn) {
                v2f bf;
                bf.x = Bsb[kb * 128 + wn + tn * 16 + ln];
                bf.y = Bsb[(kb + 1) * 128 + wn + tn * 16 + ln];
                bfrag[tn] = bf;
            }
            #pragma unroll
            for (int tm = 0; tm < 2; ++tm)
                #pragma unroll
                for (int tn = 0; tn < 2; ++tn)
                    acc[tm][tn] = __builtin_amdgcn_wmma_f32_16x16x4_f32(
                        false, afrag[tm], false, bfrag[tn],
                        (short)0, acc[tm][tn], false, false);
        }
        // no trailing barrier needed: next iteration's wait+barrier orders the
        // buffer reuse (async writes target the buffer last read at kc-1)
    }

    // epilogue: C/D layout — lane half 0 rows m..m+7, half 1 rows m+8..m+15
    #pragma unroll
    for (int tm = 0; tm < 2; ++tm) {
        #pragma unroll
        for (int tn = 0; tn < 2; ++tn) {
            const int col = n0 + wn + tn * 16 + ln;
            const float bv = bias ? bias[col] : 0.0f;
            #pragma unroll
            for (int r = 0; r < 8; ++r) {
                const int row = m0 + wm + tm * 16 + half * 8 + r;
                float val = acc[tm][tn][r] + bv;
                if (act == 1)  // exact-erf GELU
                    val = 0.5f * val * (1.0f + erff(val * 0.70710678118654752f));
                C[(size_t)row * ldc + col] = val;
            }
        }
    }
}

// ---------------------------------------------------------------------------
// Mean over tokens: gelu_out (B*NTOK, 384) -> g_mean (B, 384)
// (mean commutes with the linear W_pool2, saving the 59 GF pool2 GEMM on the
// full token axis)
// ---------------------------------------------------------------------------
__global__ void mean_tokens(const float* __restrict__ g, float* __restrict__ gm)
{
    const int b = blockIdx.x;
    const int c = threadIdx.x;            // blockDim.x == 384
    float s = 0.0f;
    for (int n = 0; n < NTOK; ++n)
        s += g[((size_t)b * NTOK + n) * POOL1C + c];
    gm[(size_t)b * POOL1C + c] = s * (1.0f / (float)NTOK);
}

// ---------------------------------------------------------------------------
// Per-batch agent attention core. One 256-thread block per batch.
// NUM_HEADS==1 => softmax over axis=1 (the head axis) is identically 1.0, so
// both blends reduce to  x * (softmax_last + 2) / 3.
// ---------------------------------------------------------------------------
__global__ __launch_bounds__(256)
void agent_attention(const float* __restrict__ qkv,     // (B*NTOK, 2304) q|k|v
                     const float* __restrict__ atraw,   // (B, 4*768)
                     const float* __restrict__ a1,      // (B,1,4,196)
                     const float* __restrict__ a2,
                     const float* __restrict__ tk1,     // (B,1,4,768)
                     const float* __restrict__ tk2,
                     float* __restrict__ rep_out,       // (B,1,4,196)
                     float* __restrict__ tok_out,       // (B,1,4,768)
                     float* __restrict__ out_pre)       // (B*NTOK, 768)
{
    __shared__ float at[AGENTS * DIMC];      // agent tokens (raw, then blended)
    __shared__ float av[AGENTS * DIMC];      // agent_v
    __shared__ float attnL[AGENTS * NTOK];   // stage-1 attention
    __shared__ float sL[AGENTS * NTOK];      // external attn blend
    __shared__ float qp[NTOK * AGENTS];      // stage-2 probabilities
    __shared__ float red[8];

    const int tid = threadIdx.x;
    const int b   = blockIdx.x;
    const float* qkvb = qkv + (size_t)b * NTOK * QKVC;

    for (int i = tid; i < AGENTS * DIMC; i += 256)
        at[i] = atraw[(size_t)b * AGENTS * DIMC + i];
    __syncthreads();

    // ---- stage 1: agent_tokens * SCALE @ k^T -> (4, 196) logits
    if (tid < NTOK) {
        const float* kr = qkvb + (size_t)tid * QKVC + DIMC;
        float c0 = 0, c1 = 0, c2 = 0, c3 = 0;
        for (int d = 0; d < DIMC; ++d) {
            const float kv = kr[d];
            c0 += at[d] * kv;            c1 += at[DIMC + d] * kv;
            c2 += at[2 * DIMC + d] * kv; c3 += at[3 * DIMC + d] * kv;
        }
        attnL[tid]            = c0 * ATTN_SCALE;
        attnL[NTOK + tid]     = c1 * ATTN_SCALE;
        attnL[2 * NTOK + tid] = c2 * ATTN_SCALE;
        attnL[3 * NTOK + tid] = c3 * ATTN_SCALE;
    }
    __syncthreads();

    // softmax over n per agent row; emit agent_rep
    if (tid < 4) {
        float m = -1e30f;
        for (int n = 0; n < NTOK; ++n) m = fmaxf(m, attnL[tid * NTOK + n]);
        red[tid] = m;
    }
    __syncthreads();
    for (int i = tid; i < AGENTS * NTOK; i += 256)
        attnL[i] = expf(attnL[i] - red[i / NTOK]);
    __syncthreads();
    if (tid < 4) {
        float s = 0.0f;
        for (int n = 0; n < NTOK; ++n) s += attnL[tid * NTOK + n];
        red[4 + tid] = 1.0f / s;
    }
    __syncthreads();
    for (int i = tid; i < AGENTS * NTOK; i += 256) {
        const float p = attnL[i] * red[4 + i / NTOK];
        attnL[i] = p;
        rep_out[(size_t)b * AGENTS * NTOK + i] = p;
    }
    __syncthreads();

    // ---- blend with softmax_n(attn_1 + attn_2); axis=1 softmax == 1
    for (int i = tid; i < AGENTS * NTOK; i += 256)
        sL[i] = a1[(size_t)b * AGENTS * NTOK + i] + a2[(size_t)b * AGENTS * NTOK + i];
    __syncthreads();
    if (tid < 4) {
        float m = -1e30f;
        for (int n = 0; n < NTOK; ++n) m = fmaxf(m, sL[tid * NTOK + n]);
        red[tid] = m;
    }
    __syncthreads();
    for (int i = tid; i < AGENTS * NTOK; i += 256)
        sL[i] = expf(sL[i] - red[i / NTOK]);
    __syncthreads();
    if (tid < 4) {
        float s = 0.0f;
        for (int n = 0; n < NTOK; ++n) s += sL[tid * NTOK + n];
        red[4 + tid] = 1.0f / s;
    }
    __syncthreads();
    for (int i = tid; i < AGENTS * NTOK; i += 256)
        attnL[i] = attnL[i] * (sL[i] * red[4 + i / NTOK] + 2.0f) * (1.0f / 3.0f);
    __syncthreads();

    // ---- agent_v = blended_attn @ v  (4, 768); thread owns 3 d-columns
    {
        float acc[AGENTS][3] = {};
        for (int n = 0; n < NTOK; ++n) {
            const float* vr = qkvb + (size_t)n * QKVC + 2 * DIMC;
            const float w0 = attnL[n],            w1 = attnL[NTOK + n];
            const float w2 = attnL[2 * NTOK + n], w3 = attnL[3 * NTOK + n];
            #pragma unroll
            for (int j = 0; j < 3; ++j) {
                const float vv = vr[tid + j * 256];
                acc[0][j] += w0 * vv; acc[1][j] += w1 * vv;
                acc[2][j] += w2 * vv; acc[3][j] += w3 * vv;
            }
        }
        #pragma unroll
        for (int a = 0; a < AGENTS; ++a)
            #pragma unroll
            for (int j = 0; j < 3; ++j)
                av[a * DIMC + tid + j * 256] = acc[a][j];
    }
    __syncthreads();

    // ---- agent token blend: at * (softmax_d(tk1+tk2) + 2) / 3
    if (tid < 4) {
        const float* t1 = tk1 + (size_t)b * AGENTS * DIMC + tid * DIMC;
        const float* t2 = tk2 + (size_t)b * AGENTS * DIMC + tid * DIMC;
        float m = -1e30f;
        for (int d = 0; d < DIMC; ++d) m = fmaxf(m, t1[d] + t2[d]);
        float s = 0.0f;
        for (int d = 0; d < DIMC; ++d) s += expf(t1[d] + t2[d] - m);
        red[tid] = m;
        red[4 + tid] = 1.0f / s;
    }
    __syncthreads();
    for (int i = tid; i < AGENTS * DIMC; i += 256) {
        const int a = i / DIMC;
        const float tv  = tk1[(size_t)b * AGENTS * DIMC + i] + tk2[(size_t)b * AGENTS * DIMC + i];
        const float smv = expf(tv - red[a]) * red[4 + a];
        const float atb = at[i] * (smv + 2.0f) * (1.0f / 3.0f);
        at[i] = atb;
        tok_out[(size_t)b * AGENTS * DIMC + i] = atb;
    }
    __syncthreads();

    // ---- stage 2: softmax_a(q*SCALE @ at_blend^T), then @ agent_v
    if (tid < NTOK) {
        const float* qr = qkvb + (size_t)tid * QKVC;
        float l0 = 0, l1 = 0, l2 = 0, l3 = 0;
        for (int d = 0; d < DIMC; ++d) {
            const float qv = qr[d];
            l0 += qv * at[d];            l1 += qv * at[DIMC + d];
            l2 += qv * at[2 * DIMC + d]; l3 += qv * at[3 * DIMC + d];
        }
        l0 *= ATTN_SCALE; l1 *= ATTN_SCALE; l2 *= ATTN_SCALE; l3 *= ATTN_SCALE;
        const float m  = fmaxf(fmaxf(l0, l1), fmaxf(l2, l3));
        const float e0 = expf(l0 - m), e1 = expf(l1 - m);
        const float e2 = expf(l2 - m), e3 = expf(l3 - m);
        const float inv = 1.0f / (e0 + e1 + e2 + e3);
        qp[tid * 4 + 0] = e0 * inv; qp[tid * 4 + 1] = e1 * inv;
        qp[tid * 4 + 2] = e2 * inv; qp[tid * 4 + 3] = e3 * inv;
    }
    __syncthreads();
    for (int i = tid; i < NTOK * DIMC; i += 256) {
        const int n = i / DIMC, d = i - n * DIMC;
        const float o = qp[n * 4 + 0] * av[d]
                      + qp[n * 4 + 1] * av[DIMC + d]
                      + qp[n * 4 + 2] * av[2 * DIMC + d]
                      + qp[n * 4 + 3] * av[3 * DIMC + d];
        out_pre[((size_t)b * NTOK + n) * DIMC + d] = o;
    }
}

// ---------------------------------------------------------------------------
extern "C" void kernel_launch(void* const* d_in, const int* in_sizes, int n_in,
                              void* d_out, int out_size, void* d_ws, size_t ws_size,
                              hipStream_t stream)
{
    (void)in_sizes; (void)n_in; (void)out_size; (void)ws_size;

    const float* x     = (const float*)d_in[0];
    const float* attn1 = (const float*)d_in[1];
    const float* attn2 = (const float*)d_in[2];
    const float* tk1   = (const float*)d_in[3];
    const float* tk2   = (const float*)d_in[4];
    const float* Wqkv  = (const float*)d_in[5];
    const float* Wp1   = (const float*)d_in[6];
    const float* bp1   = (const float*)d_in[7];
    const float* Wp2   = (const float*)d_in[8];
    const float* bp2   = (const float*)d_in[9];
    const float* Wpr   = (const float*)d_in[10];
    const float* bpr   = (const float*)d_in[11];

    float* out = (float*)d_out;
    float* ws  = (float*)d_ws;

    // workspace layout (floats)
    float* qkv    = ws;                                        // 25088 x 2304
    float* gelu   = qkv   + (size_t)MROWS * QKVC;              // 25088 x 384
    float* gmean  = gelu  + (size_t)MROWS * POOL1C;            // 128 x 384
    float* atraw  = gmean + (size_t)BATCH * POOL1C;            // 128 x 3072
    float* outpre = atraw + (size_t)BATCH * POOL2C;            // 25088 x 768

    // output layout: out (B,N,C) | agent_rep (B,1,4,196) | agent_tokens (B,1,4,768)
    float* out_main = out;
    float* rep_out  = out + (size_t)MROWS * DIMC;
    float* tok_out  = rep_out + (size_t)BATCH * AGENTS * NTOK;

    const dim3 blk(256);

    // qkv = x @ W_qkv                       (25088,768)@(768,2304)
    gemm_f32_wmma<<<dim3(QKVC / 128, MROWS / 64), blk, 0, stream>>>(
        x, DIMC, Wqkv, QKVC, nullptr, qkv, QKVC, MROWS, QKVC, DIMC, 0);

    // gelu = GELU(q @ W_pool1 + b_pool1)    (25088,768)@(768,384); q = qkv[:, :768]
    gemm_f32_wmma<<<dim3(POOL1C / 128, MROWS / 64), blk, 0, stream>>>(
        qkv, QKVC, Wp1, POOL1C, bp1, gelu, POOL1C, MROWS, POOL1C, DIMC, 1);

    // g_mean = mean over tokens             (128,384)
    mean_tokens<<<dim3(BATCH), dim3(POOL1C), 0, stream>>>(gelu, gmean);

    // agent_tokens_raw = g_mean @ W_pool2 + b_pool2   (128,384)@(384,3072)
    gemm_f32_wmma<<<dim3(POOL2C / 128, BATCH / 64), blk, 0, stream>>>(
        gmean, POOL1C, Wp2, POOL2C, bp2, atraw, POOL2C, BATCH, POOL2C, POOL1C, 0);

    // per-batch agent attention core
    agent_attention<<<dim3(BATCH), blk, 0, stream>>>(
        qkv, atraw, attn1, attn2, tk1, tk2, rep_out, tok_out, outpre);

    // out = out_pre @ W_proj + b_proj       (25088,768)@(768,768)
    gemm_f32_wmma<<<dim3(DIMC / 128, MROWS / 64), blk, 0, stream>>>(
        outpre, DIMC, Wpr, DIMC, bpr, out_main, DIMC, MROWS, DIMC, DIMC, 0);
}